// ProbabilisticSurfaceDistanceLoss_44220983280150
// MI455X (gfx1250) — compile-verified
//
#include <hip/hip_runtime.h>
#include <math.h>

typedef __attribute__((ext_vector_type(2))) float v2f;
typedef __attribute__((ext_vector_type(8))) float v8f;

// Exact-instruction float mins (distances are never NaN, min_num is safe).
#define FMIN3_ACC(acc, x, y) \
    asm("v_min3_num_f32 %0, %0, %1, %2" : "+v"(acc) : "v"(x), "v"(y))
#define FMIN_ACC(acc, x) \
    asm("v_min_num_f32 %0, %0, %1" : "+v"(acc) : "v"(x))

// ---------------------------------------------------------------------------
// Prep: build operand arrays for the WMMA distance sweeps.
//  A-side (query points, float4 = x,y,z,|p|^2):
//    sbary4 : simplified-face barycenters [G]
//    pts4   : sampled surface points      [T = G*S]
//  B-side (targets, WMMA-lane-interleaved v2f, 2 entries per target):
//    pre[(n>>4)*32 + (n&15)]      = {-2x_n, -2y_n}   (lanes 0-15 : K0,K1)
//    pre[(n>>4)*32 + 16 + (n&15)] = {-2z_n, |v_n|^2} (lanes 16-31: K2,K3)
//  Pad targets get q = 1e30 so they never win a min-distance.
// ---------------------------------------------------------------------------
__global__ void prep_kernel(const float* __restrict__ overts,
                            const int*   __restrict__ ofaces,
                            const float* __restrict__ sverts,
                            const int*   __restrict__ sfaces,
                            const float* __restrict__ r1,
                            const float* __restrict__ r2,
                            v2f*    __restrict__ vpre,
                            v2f*    __restrict__ obpre,
                            float4* __restrict__ sbary4,
                            float4* __restrict__ pts4,
                            int N, int Npad, int F, int Fpad, int G, int S, int T)
{
    int i = blockIdx.x * blockDim.x + threadIdx.x;

    if (i < Npad) {
        float x = 0.f, y = 0.f, z = 0.f, q = 1.0e30f;
        if (i < N) {
            x = overts[3*i+0]; y = overts[3*i+1]; z = overts[3*i+2];
            q = x*x + y*y + z*z;
        }
        int base = (i >> 4) * 32 + (i & 15);
        v2f lo;  lo.x  = -2.0f * x; lo.y  = -2.0f * y;
        v2f hiv; hiv.x = -2.0f * z; hiv.y = q;
        vpre[base]      = lo;
        vpre[base + 16] = hiv;
    }
    if (i < Fpad) {
        float x = 0.f, y = 0.f, z = 0.f, q = 1.0e30f;
        if (i < F) {
            int a = ofaces[3*i+0], b = ofaces[3*i+1], c = ofaces[3*i+2];
            x = (overts[3*a+0] + overts[3*b+0] + overts[3*c+0]) * (1.0f/3.0f);
            y = (overts[3*a+1] + overts[3*b+1] + overts[3*c+1]) * (1.0f/3.0f);
            z = (overts[3*a+2] + overts[3*b+2] + overts[3*c+2]) * (1.0f/3.0f);
            q = x*x + y*y + z*z;
        }
        int base = (i >> 4) * 32 + (i & 15);
        v2f lo;  lo.x  = -2.0f * x; lo.y  = -2.0f * y;
        v2f hiv; hiv.x = -2.0f * z; hiv.y = q;
        obpre[base]      = lo;
        obpre[base + 16] = hiv;
    }
    if (i < G) {
        int a = sfaces[3*i+0], b = sfaces[3*i+1], c = sfaces[3*i+2];
        float x = (sverts[3*a+0] + sverts[3*b+0] + sverts[3*c+0]) * (1.0f/3.0f);
        float y = (sverts[3*a+1] + sverts[3*b+1] + sverts[3*c+1]) * (1.0f/3.0f);
        float z = (sverts[3*a+2] + sverts[3*b+2] + sverts[3*c+2]) * (1.0f/3.0f);
        sbary4[i] = make_float4(x, y, z, x*x + y*y + z*z);
    }
    if (i < T) {
        int g = i / S;
        int a = sfaces[3*g+0], b = sfaces[3*g+1], c = sfaces[3*g+2];
        float sr = sqrtf(r1[i]);
        float u = 1.0f - sr;
        float v = sr * (1.0f - r2[i]);
        float w = sr * r2[i];
        float x = u*sverts[3*a+0] + v*sverts[3*b+0] + w*sverts[3*c+0];
        float y = u*sverts[3*a+1] + v*sverts[3*b+1] + w*sverts[3*c+1];
        float z = u*sverts[3*a+2] + v*sverts[3*b+2] + w*sverts[3*c+2];
        pts4[i] = make_float4(x, y, z, x*x + y*y + z*z);
    }
}

// ---------------------------------------------------------------------------
// Core: one wave owns two 16-row M-tiles of query points A; sweeps all 16-col
// N-tiles of pre-built B with V_WMMA_F32_16X16X4_F32 and inline-zero C:
//   A(16x4) = (x, y, z, 1)      per point m
//   B(4x16) = (-2x,-2y,-2z,q)   per target n (prebuilt, one b64 load/lane)
//   D = A*B + 0 = -2 p.v + |v|^2          (|p|^2 added in the cold epilogue:
//   min_n(|p|^2 + D) = |p|^2 + min_n D, so C stays inline 0 -> no C tuple,
//   no per-iteration register copies.)
// N-loop manually unrolled x2 with explicit v_min3_num_f32 accumulation.
// Then xor-shuffle min within each 16-lane half.
// rm[mt][j]: lane 0  -> min over n of D for m = mbase+mt*16+j,
//            lane 16 -> same for m = mbase+mt*16+j+8.
// ---------------------------------------------------------------------------
__device__ __forceinline__ void wmma_min_d2(const float4* __restrict__ A,
                                            int mbase, int Mtot,
                                            const v2f* __restrict__ Bpre,
                                            int ntiles, float rm[2][8])
{
    const int  lane = threadIdx.x & 31;
    const bool hi   = lane >= 16;
    const int  l15  = lane & 15;

    v2f a[2];
#pragma unroll
    for (int mt = 0; mt < 2; ++mt) {
        int mi = mbase + mt * 16 + l15;
        if (mi >= Mtot) mi = Mtot - 1;
        float4 ap = A[mi];
        a[mt].x = hi ? ap.z : ap.x;   // lanes 0-15: {x,y}; 16-31: {z,1}
        a[mt].y = hi ? 1.0f : ap.y;
#pragma unroll
        for (int j = 0; j < 8; ++j) rm[mt][j] = 3.0e38f;
    }
    const v8f czero = {};             // lowers to inline-constant SRC2 = 0

    const v2f* __restrict__ bp = Bpre + lane;
    int t = 0;
    for (; t + 1 < ntiles; t += 2) {
        v2f b0 = bp[t * 32];          // global_load_b64
        v2f b1 = bp[t * 32 + 32];     // global_load_b64 (offset 256B)
        v8f d00 = __builtin_amdgcn_wmma_f32_16x16x4_f32(
            false, a[0], false, b0, (short)0, czero, false, false);
        v8f d01 = __builtin_amdgcn_wmma_f32_16x16x4_f32(
            false, a[0], false, b1, (short)0, czero, false, false);
        v8f d10 = __builtin_amdgcn_wmma_f32_16x16x4_f32(
            false, a[1], false, b0, (short)0, czero, false, false);
        v8f d11 = __builtin_amdgcn_wmma_f32_16x16x4_f32(
            false, a[1], false, b1, (short)0, czero, false, false);
#pragma unroll
        for (int j = 0; j < 8; ++j) {
            FMIN3_ACC(rm[0][j], d00[j], d01[j]);
            FMIN3_ACC(rm[1][j], d10[j], d11[j]);
        }
    }
    for (; t < ntiles; ++t) {         // remainder (ntiles odd)
        v2f b = bp[t * 32];
        v8f d0 = __builtin_amdgcn_wmma_f32_16x16x4_f32(
            false, a[0], false, b, (short)0, czero, false, false);
        v8f d1 = __builtin_amdgcn_wmma_f32_16x16x4_f32(
            false, a[1], false, b, (short)0, czero, false, false);
#pragma unroll
        for (int j = 0; j < 8; ++j) {
            FMIN_ACC(rm[0][j], d0[j]);
            FMIN_ACC(rm[1][j], d1[j]);
        }
    }

#pragma unroll
    for (int mt = 0; mt < 2; ++mt) {
#pragma unroll
        for (int j = 0; j < 8; ++j) {
            float v = rm[mt][j];
            v = fminf(v, __shfl_xor(v, 1, 32));
            v = fminf(v, __shfl_xor(v, 2, 32));
            v = fminf(v, __shfl_xor(v, 4, 32));
            v = fminf(v, __shfl_xor(v, 8, 32));
            rm[mt][j] = v;
        }
    }
}

// Forward: per-wave partial of  sum_m p_m*min_d(m) + 1e-4*(1-p_m)
__global__ void fwd_min_kernel(const float4* __restrict__ sbary4, int G,
                               const v2f*    __restrict__ obpre, int ntiles,
                               const float*  __restrict__ probs,
                               float* __restrict__ fwd_part)
{
    int mbase = blockIdx.x * 32;
    float rm[2][8];
    wmma_min_d2(sbary4, mbase, G, obpre, ntiles, rm);
    int lane = threadIdx.x & 31;
    float local = 0.0f;
    if (lane == 0 || lane == 16) {
#pragma unroll
        for (int mt = 0; mt < 2; ++mt) {
            int moff = mbase + mt * 16 + (lane ? 8 : 0);
#pragma unroll
            for (int j = 0; j < 8; ++j) {
                int m = moff + j;
                if (m < G) {
                    float d2 = rm[mt][j] + sbary4[m].w;   // add |p|^2 back
                    float dmin = sqrtf(fmaxf(d2, 1e-12f));
                    float p = probs[m];
                    local += p * dmin + 1.0e-4f * (1.0f - p);
                }
            }
        }
    }
    float o = __shfl(local, 16, 32);
    if (lane == 0) fwd_part[blockIdx.x] = local + o;
}

// Reverse: per sampled point min distance to original vertices.
__global__ void rev_min_kernel(const float4* __restrict__ pts4, int T,
                               const v2f*    __restrict__ vpre, int ntiles,
                               float* __restrict__ min_rev)
{
    int mbase = blockIdx.x * 32;
    float rm[2][8];
    wmma_min_d2(pts4, mbase, T, vpre, ntiles, rm);
    int lane = threadIdx.x & 31;
    if (lane == 0 || lane == 16) {
#pragma unroll
        for (int mt = 0; mt < 2; ++mt) {
            int moff = mbase + mt * 16 + (lane ? 8 : 0);
#pragma unroll
            for (int j = 0; j < 8; ++j) {
                int m = moff + j;
                if (m < T) {
                    float d2 = rm[mt][j] + pts4[m].w;     // add |p|^2 back
                    min_rev[m] = sqrtf(fmaxf(d2, 1e-12f));
                }
            }
        }
    }
}

// Final deterministic reduction: rev sum, rev max, fwd partial sum -> scalar.
__global__ void final_kernel(const float* __restrict__ min_rev,
                             const float* __restrict__ probs,
                             const float* __restrict__ fwd_part, int nfwd,
                             int T, int S, float* __restrict__ out)
{
    __shared__ float s_rev[256];
    __shared__ float s_fwd[256];
    __shared__ float s_max[256];
    int t = threadIdx.x;
    float sum = 0.0f, mx = 0.0f, fs = 0.0f;
    for (int i = t; i < T; i += 256) {
        float d = min_rev[i];
        sum += probs[i / S] * d;
        mx = fmaxf(mx, d);
    }
    for (int i = t; i < nfwd; i += 256) fs += fwd_part[i];
    s_rev[t] = sum; s_fwd[t] = fs; s_max[t] = mx;
    __syncthreads();
    for (int off = 128; off > 0; off >>= 1) {
        if (t < off) {
            s_rev[t] += s_rev[t + off];
            s_fwd[t] += s_fwd[t + off];
            s_max[t]  = fmaxf(s_max[t], s_max[t + off]);
        }
        __syncthreads();
    }
    if (t == 0)
        out[0] = s_fwd[0] + s_rev[0] * 0.1f / (s_max[0] + 1.0e-8f);
}

extern "C" void kernel_launch(void* const* d_in, const int* in_sizes, int n_in,
                              void* d_out, int out_size, void* d_ws, size_t ws_size,
                              hipStream_t stream)
{
    const float* overts = (const float*)d_in[0];
    const int*   ofaces = (const int*)  d_in[1];
    const float* sverts = (const float*)d_in[2];
    const int*   sfaces = (const int*)  d_in[3];
    const float* probs  = (const float*)d_in[4];
    const float* r1     = (const float*)d_in[5];
    const float* r2     = (const float*)d_in[6];

    const int N = in_sizes[0] / 3;          // 16384 original vertices
    const int F = in_sizes[1] / 3;          // 16384 original faces
    const int G = in_sizes[4];              // 8192 simplified faces
    const int S = in_sizes[5] / G;          // 16 samples per face
    const int T = G * S;                    // 131072 sampled points

    const int Npad = (N + 15) & ~15;
    const int Fpad = (F + 15) & ~15;

    float* ws = (float*)d_ws;
    v2f*    vpre   = (v2f*)ws;                                 // Npad*2 v2f
    v2f*    obpre  = (v2f*)(ws + 4 * Npad);                    // Fpad*2 v2f
    float4* sbary4 = (float4*)(ws + 4 * (Npad + Fpad));        // G
    float4* pts4   = (float4*)(ws + 4 * (Npad + Fpad + G));    // T
    float*  min_rev  = ws + 4 * (Npad + Fpad + G + T);         // T
    float*  fwd_part = min_rev + T;                            // (G+31)/32

    int prep_n = Npad;
    if (Fpad > prep_n) prep_n = Fpad;
    if (T    > prep_n) prep_n = T;
    prep_kernel<<<(prep_n + 255) / 256, 256, 0, stream>>>(
        overts, ofaces, sverts, sfaces, r1, r2,
        vpre, obpre, sbary4, pts4, N, Npad, F, Fpad, G, S, T);

    const int fwd_blocks = (G + 31) / 32;   // 256 waves, 2 M-tiles each
    fwd_min_kernel<<<fwd_blocks, 32, 0, stream>>>(
        sbary4, G, obpre, Fpad / 16, probs, fwd_part);

    const int rev_blocks = (T + 31) / 32;   // 4096 waves, 2 M-tiles each
    rev_min_kernel<<<rev_blocks, 32, 0, stream>>>(
        pts4, T, vpre, Npad / 16, min_rev);

    final_kernel<<<1, 256, 0, stream>>>(
        min_rev, probs, fwd_part, fwd_blocks, T, S, (float*)d_out);
}